// ResiduesEncode_21251498181156
// MI455X (gfx1250) — compile-verified
//
#include <hip/hip_runtime.h>

// ResiduesEncode for MI455X (gfx1250): batched 16x24 (f32) x 24x16 GEMM per
// residue via V_WMMA_F32_16X16X4_F32 (fp32 matrix pipe, exact precision).
//
// Shapes (hardcoded from reference): B=64, VDIM=16, MAXN=24, T=20.
// Per residue r: x_v[b,v,d] = sum_n W[seq[r],v,n] * delta[n,b,d]
//   -> M=16(v), K=24(n) in 6 steps of K=4, N=16 batch elems; tiles over
//      (d in 0..2) x (btile in 0..3) = 12 wave32 tiles per workgroup.

typedef __attribute__((ext_vector_type(2))) float v2f;
typedef __attribute__((ext_vector_type(8))) float v8f;

#define BATCH 64
#define VDIM  16
#define MAXN  24
#define NTHREADS 384          // 12 waves (wave32)
#define DPAD  72              // padded b-stride for delta: 2*72 mod 64 = 16 -> halves hit disjoint banks
#define XVPAD 49              // odd stride -> conflict-free D staging

__launch_bounds__(NTHREADS)
__global__ void residues_encode_kernel(const float* __restrict__ x,      // (B, A, 3)
                                       const float* __restrict__ W,      // (T, 16, 24)
                                       const int*   __restrict__ seq,    // (R)
                                       const int*   __restrict__ ridx,   // (R)
                                       const int*   __restrict__ natoms, // (T)
                                       float* __restrict__ out_root,     // (B, R, 3)
                                       float* __restrict__ out_xv,       // (B, R, 16, 3)
                                       int A, int R)
{
    __shared__ float lds_root[3][BATCH];
    __shared__ float lds_delta[3][MAXN][DPAD];   // 5184 floats; reused for D staging (needs 64*49=3136)

    const int r = blockIdx.x;
    const int t = threadIdx.x;

    const int sr = seq[r];
    const int ri = ridx[r];
    const int na = natoms[sr];

    // ---- Phase 1: root = x[:, ri+1, :]  (also emitted as first output) ----
    if (t < BATCH * 3) {
        const int b = t / 3, d = t % 3;
        const float rv = x[((long)b * A + (ri + 1)) * 3 + d];
        lds_root[d][b] = rv;
        out_root[((long)b * R + r) * 3 + d] = rv;
    }
    __syncthreads();

    // ---- Phase 2: delta[n,b,d] = (x[b, ri+n, d] - root[b,d]) * (n < na) ----
    // e = b*72 + n*3 + d : consecutive threads read contiguous 72-float runs of x.
    for (int e = t; e < BATCH * MAXN * 3; e += NTHREADS) {
        const int b = e / (MAXN * 3);
        const int inner = e % (MAXN * 3);
        const int n = inner / 3, d = inner % 3;
        float v = 0.0f;
        if (n < na) {
            int atom = ri + n;
            if (atom > A - 1) atom = A - 1;   // reference clamp
            v = x[((long)b * A + atom) * 3 + d] - lds_root[d][b];
        }
        lds_delta[d][n][b] = v;
    }
    __syncthreads();

    // ---- Phase 3: 12 waves, one 16x16 output tile each; 6 K-steps of wmma f32 16x16x4 ----
    const int wave  = t >> 5;
    const int lane  = t & 31;
    const int half  = lane >> 4;      // K sub-offset select (A/B layouts: +2 for lanes 16..31)
    const int col   = lane & 15;      // A: M index; B/D: N (batch) index
    const int d     = wave >> 2;      // 0..2
    const int btile = wave & 3;       // 0..3
    const int b     = btile * 16 + col;

    // A fragment source: a[0] = W[sr][m][4k + 2*half], a[1] = next (8B-aligned pair -> b64 load)
    const float* Wr = W + ((long)(sr * VDIM + col) * MAXN + 2 * half);

    v8f acc = {};
    #pragma unroll
    for (int k = 0; k < 6; ++k) {
        v2f a;
        a[0] = Wr[4 * k];
        a[1] = Wr[4 * k + 1];
        v2f bb;
        bb[0] = lds_delta[d][4 * k + 2 * half + 0][b];
        bb[1] = lds_delta[d][4 * k + 2 * half + 1][b];
        // D = A(16x4 f32) * B(4x16 f32) + C, full-wave, fp32 accumulate
        acc = __builtin_amdgcn_wmma_f32_16x16x4_f32(
            /*neg_a=*/false, a, /*neg_b=*/false, bb,
            /*c_mod=*/(short)0, acc, /*reuse_a=*/false, /*reuse_b=*/false);
    }
    __syncthreads();   // all waves done reading delta; safe to reuse LDS

    // ---- Stage D through LDS (reuse delta space) for coalesced writeout ----
    float* xv = &lds_delta[0][0][0];
    #pragma unroll
    for (int p = 0; p < 8; ++p) {
        const int v = p + 8 * half;                 // D layout: VGPR p -> M = p (+8 for lanes 16..31)
        xv[b * XVPAD + v * 3 + d] = acc[p];
    }
    __syncthreads();

    // ---- Phase 4: coalesced store of x_v (48 contiguous floats per batch elem) ----
    for (int e = t; e < BATCH * VDIM * 3; e += NTHREADS) {
        const int b2 = e / (VDIM * 3);
        const int j  = e % (VDIM * 3);
        out_xv[((long)b2 * R + r) * (VDIM * 3) + j] = xv[b2 * XVPAD + j];
    }
}

extern "C" void kernel_launch(void* const* d_in, const int* in_sizes, int n_in,
                              void* d_out, int out_size, void* d_ws, size_t ws_size,
                              hipStream_t stream) {
    const float* x      = (const float*)d_in[0];   // (B, A, 3) f32
    const float* W      = (const float*)d_in[1];   // (T, 16, 24) f32
    const int*   seq    = (const int*)d_in[2];     // (R) i32
    const int*   ridx   = (const int*)d_in[3];     // (R) i32
    const int*   natoms = (const int*)d_in[4];     // (T) i32

    const int R = in_sizes[2];
    const int A = in_sizes[0] / (BATCH * 3);

    float* out_root = (float*)d_out;                       // (B, R, 3)
    float* out_xv   = out_root + (size_t)BATCH * R * 3;    // (B, R, 16, 3)

    residues_encode_kernel<<<R, NTHREADS, 0, stream>>>(x, W, seq, ridx, natoms,
                                                       out_root, out_xv, A, R);
}